// MLP_36026185678951
// MI455X (gfx1250) — compile-verified
//
#include <hip/hip_runtime.h>
#include <math.h>

typedef __attribute__((ext_vector_type(2))) float v2f;
typedef __attribute__((ext_vector_type(8))) float v8f;

#define BLK            256
#define ROWS_PER_BLOCK 256
#define K1   784
#define N1   12
#define N1P  16   // padded N for the 16-wide WMMA tile

// Small-weight LDS offsets (floats)
#define OFF_W2 0     // 12*10 = 120
#define OFF_B2 120   // 10
#define OFF_W3 130   // 10*8 = 80
#define OFF_B3 210   // 8
#define OFF_W4 218   // 8*6 = 48
#define OFF_B4 266   // 6
#define OFF_W5 272   // 6*4 = 24
#define OFF_B5 296   // 4
#define OFF_W6 300   // 4*10 = 40
#define OFF_B6 340   // 10
#define SW_TOTAL 350

__global__ __launch_bounds__(BLK) void mlp6_wmma_kernel(
    const float* __restrict__ x,
    const float* __restrict__ W1, const float* __restrict__ b1,
    const float* __restrict__ W2, const float* __restrict__ b2,
    const float* __restrict__ W3, const float* __restrict__ b3,
    const float* __restrict__ W4, const float* __restrict__ b4,
    const float* __restrict__ W5, const float* __restrict__ b5,
    const float* __restrict__ W6, const float* __restrict__ b6,
    float* __restrict__ out, int B)
{
    // LDS: padded W1 (784x16), h1 staging (256 rows x stride 13), small weights
    __shared__ float sW1[K1 * N1P];                 // 50176 B
    __shared__ float sH1[ROWS_PER_BLOCK * 13];      // 13312 B
    __shared__ float sSW[SW_TOTAL];                 //  1400 B

    const int t = threadIdx.x;

    // ---- Stage W1 into LDS, zero-padded to 16 columns ----
    for (int idx = t; idx < K1 * N1P; idx += BLK) {
        const int row = idx >> 4;
        const int col = idx & 15;
        sW1[idx] = (col < N1) ? W1[row * N1 + col] : 0.0f;
    }
    // ---- Stage small tail weights into LDS ----
    for (int idx = t; idx < SW_TOTAL; idx += BLK) {
        float v;
        if      (idx < OFF_B2) v = W2[idx - OFF_W2];
        else if (idx < OFF_W3) v = b2[idx - OFF_B2];
        else if (idx < OFF_B3) v = W3[idx - OFF_W3];
        else if (idx < OFF_W4) v = b3[idx - OFF_B3];
        else if (idx < OFF_B4) v = W4[idx - OFF_W4];
        else if (idx < OFF_W5) v = b4[idx - OFF_B4];
        else if (idx < OFF_B5) v = W5[idx - OFF_W5];
        else if (idx < OFF_W6) v = b5[idx - OFF_B5];
        else if (idx < OFF_B6) v = W6[idx - OFF_W6];
        else                   v = b6[idx - OFF_B6];
        sSW[idx] = v;
    }
    __syncthreads();

    const int lane  = t & 31;
    const int wave  = t >> 5;
    const int n     = lane & 15;        // column within 16-wide tile
    const int kSel  = (lane >> 4) << 1; // lanes 0-15: K+0/+1 ; lanes 16-31: K+2/+3
    const int blockBase = blockIdx.x * ROWS_PER_BLOCK;

    const float bias1 = (n < N1) ? b1[n] : 0.0f;

    const size_t off_out = 0;                    // logits [B,10]
    const size_t off_h1  = (size_t)B * 10;       // [B,12]
    const size_t off_h2  = (size_t)B * 22;       // [B,10]
    const size_t off_h3  = (size_t)B * 32;       // [B,8]
    const size_t off_h4  = (size_t)B * 40;       // [B,6]
    const size_t off_h5  = (size_t)B * 46;       // [B,4]

    // ============ Phase 1: fc1 via V_WMMA_F32_16X16X4_F32 ============
    // Each wave owns two 16-row tiles, interleaved in one k-loop so that
    // 8 global b64 loads are in flight per unrolled step and the B operand
    // (LDS) is shared by the back-to-back WMMAs.
    const int m0 = blockBase + wave * 32;
    if (m0 + 32 <= B) {                                  // wave-uniform guard
        const float* xrow0 = x + (size_t)(m0 +      n) * K1 + kSel;
        const float* xrow1 = x + (size_t)(m0 + 16 + n) * K1 + kSel;
        const float* bcol  = sW1 + (size_t)kSel * N1P + n;

        v8f c0 = {0.f, 0.f, 0.f, 0.f, 0.f, 0.f, 0.f, 0.f};
        v8f c1 = {0.f, 0.f, 0.f, 0.f, 0.f, 0.f, 0.f, 0.f};

        #pragma unroll 4
        for (int k = 0; k < K1; k += 4) {
            // A operands: 2 consecutive floats per lane (8B aligned) per tile
            v2f a0 = *(const v2f*)(xrow0 + k);
            v2f a1 = *(const v2f*)(xrow1 + k);
            // Shared B operand: 4x16 fp32 tile from LDS (same half-wave K split)
            v2f bm;
            bm.x = bcol[(size_t)k * N1P];
            bm.y = bcol[(size_t)k * N1P + N1P];
            c0 = __builtin_amdgcn_wmma_f32_16x16x4_f32(
                     false, a0, false, bm, (short)0, c0, false, false);
            c1 = __builtin_amdgcn_wmma_f32_16x16x4_f32(
                     false, a1, false, bm, (short)0, c1, false, false);
        }

        // D layout: VGPR v -> M = v (lanes 0-15) / v+8 (lanes 16-31), N = lane%16
        const int mBase = (lane >> 4) * 8;
        #pragma unroll
        for (int v = 0; v < 8; ++v) {
            const int r0 = m0 + mBase + v;
            const int r1 = r0 + 16;
            if (n < N1) {
                const float h0 = tanhf(c0[v] + bias1);
                const float h1v = tanhf(c1[v] + bias1);
                out[off_h1 + (size_t)r0 * N1 + n] = h0;
                out[off_h1 + (size_t)r1 * N1 + n] = h1v;
                sH1[(r0 - blockBase) * 13 + n] = h0;
                sH1[(r1 - blockBase) * 13 + n] = h1v;
            }
        }
    }
    __syncthreads();

    // ============ Phase 2: tail layers, one row per thread ============
    const int r = blockBase + t;
    if (r < B) {
        float h1[12], h2[10], h3[8], h4[6], h5[4];

        #pragma unroll
        for (int i = 0; i < 12; ++i) h1[i] = sH1[t * 13 + i];

        // fc2: 12 -> 10, tanh
        #pragma unroll
        for (int j = 0; j < 10; ++j) {
            float s = sSW[OFF_B2 + j];
            #pragma unroll
            for (int i = 0; i < 12; ++i) s = fmaf(h1[i], sSW[OFF_W2 + i * 10 + j], s);
            h2[j] = tanhf(s);
            out[off_h2 + (size_t)r * 10 + j] = h2[j];
        }
        // fc3: 10 -> 8, tanh
        #pragma unroll
        for (int j = 0; j < 8; ++j) {
            float s = sSW[OFF_B3 + j];
            #pragma unroll
            for (int i = 0; i < 10; ++i) s = fmaf(h2[i], sSW[OFF_W3 + i * 8 + j], s);
            h3[j] = tanhf(s);
            out[off_h3 + (size_t)r * 8 + j] = h3[j];
        }
        // fc4: 8 -> 6, tanh
        #pragma unroll
        for (int j = 0; j < 6; ++j) {
            float s = sSW[OFF_B4 + j];
            #pragma unroll
            for (int i = 0; i < 8; ++i) s = fmaf(h3[i], sSW[OFF_W4 + i * 6 + j], s);
            h4[j] = tanhf(s);
            out[off_h4 + (size_t)r * 6 + j] = h4[j];
        }
        // fc5: 6 -> 4, tanh
        #pragma unroll
        for (int j = 0; j < 4; ++j) {
            float s = sSW[OFF_B5 + j];
            #pragma unroll
            for (int i = 0; i < 6; ++i) s = fmaf(h4[i], sSW[OFF_W5 + i * 4 + j], s);
            h5[j] = tanhf(s);
            out[off_h5 + (size_t)r * 4 + j] = h5[j];
        }
        // fc6: 4 -> 10, no activation (logits)
        #pragma unroll
        for (int j = 0; j < 10; ++j) {
            float s = sSW[OFF_B6 + j];
            #pragma unroll
            for (int i = 0; i < 4; ++i) s = fmaf(h5[i], sSW[OFF_W6 + i * 10 + j], s);
            out[off_out + (size_t)r * 10 + j] = s;
        }
    }
}

extern "C" void kernel_launch(void* const* d_in, const int* in_sizes, int n_in,
                              void* d_out, int out_size, void* d_ws, size_t ws_size,
                              hipStream_t stream) {
    const float* x  = (const float*)d_in[0];
    const float* W1 = (const float*)d_in[1];
    const float* b1 = (const float*)d_in[2];
    const float* W2 = (const float*)d_in[3];
    const float* b2 = (const float*)d_in[4];
    const float* W3 = (const float*)d_in[5];
    const float* b3 = (const float*)d_in[6];
    const float* W4 = (const float*)d_in[7];
    const float* b4 = (const float*)d_in[8];
    const float* W5 = (const float*)d_in[9];
    const float* b5 = (const float*)d_in[10];
    const float* W6 = (const float*)d_in[11];
    const float* b6 = (const float*)d_in[12];

    const int B = in_sizes[0] / K1;            // 65536
    const int grid = (B + ROWS_PER_BLOCK - 1) / ROWS_PER_BLOCK;

    mlp6_wmma_kernel<<<grid, BLK, 0, stream>>>(
        x, W1, b1, W2, b2, W3, b3, W4, b4, W5, b5, W6, b6,
        (float*)d_out, B);
}